// pFF_channelwise_46583215293130
// MI455X (gfx1250) — compile-verified
//
#include <hip/hip_runtime.h>
#include <math.h>

// MI455X / gfx1250, wave32. fp32 WMMA (V_WMMA_F32_16X16X4_F32) for the conv GEMM.

typedef __attribute__((ext_vector_type(2))) float v2f;
typedef __attribute__((ext_vector_type(8))) float v8f;

#define HH 128
#define WW 128
#define HW (HH * WW)

__device__ __forceinline__ int reflect_idx(int i, int n) {
  if (i < 0) i = -i;
  if (i >= n) i = 2 * n - 2 - i;
  return i;
}

// ---------------------------------------------------------------------------
// Kernel 0: pack conv_w [18][64][3][3] into WMMA A-fragment order, M padded
// to 32 with zeros.  Layout: float2 [kb=0..143][mt=0..1][lane=0..31],
// where lane l of tile mt holds A[m = (l&15)+16*mt][k = 4*kb + 2*(l>>4) + v]
// in component v (matches the ISA 32-bit 16x4 A layout).
// K ordering: k = kpos*64 + ic  (kpos = kh*3+kw), so a 4-wide K block never
// straddles a kernel tap.
// ---------------------------------------------------------------------------
__global__ void pack_w(const float* __restrict__ cw, float* __restrict__ wp) {
  int t = blockIdx.x * blockDim.x + threadIdx.x;
  if (t >= 144 * 2 * 32 * 2) return;
  int v = t & 1;
  int lane = (t >> 1) & 31;
  int mt = (t >> 6) & 1;
  int kb = t >> 7;
  int m = (lane & 15) + 16 * mt;
  int hi = lane >> 4;
  int k = kb * 4 + 2 * hi + v;
  int kpos = k >> 6;      // kernel tap 0..8
  int ic = k & 63;        // input channel
  float val = 0.0f;
  if (m < 18) val = cw[(m * 64 + ic) * 9 + kpos];
  wp[t] = val;
}

// ---------------------------------------------------------------------------
// Kernel 1: conv(3x3, reflect) + grouped softmax -> ff workspace
//   grid = (H, B), block = 256 (8 waves); wave wv owns pixels [wv*16, wv*16+16)
//   of row h.  Output layout ffw[b][s=c%2][k=0..8][h][w]  (consumer order).
// ---------------------------------------------------------------------------
__global__ __launch_bounds__(256) void conv_softmax(
    const float* __restrict__ feat, const float* __restrict__ bias,
    const float* __restrict__ wp, float* __restrict__ ffw) {
  // Feature slab: 64 ch x 3 rows x 130 cols, interleaved by ic pairs so a
  // lane's B fragment (ic even, ic+1) is a single b64 LDS load.
  // float index: ((ic>>1)*3 + r)*264 + 2*wcol + (ic&1); row pitch 132 float2.
  __shared__ float xs[64 * 3 * 132];          // 101,376 B
  __shared__ float zb[8 * 16 * 19];           //   9,728 B (stride 19: no bank conflicts)

  const int h = blockIdx.x;
  const int b = blockIdx.y;
  const int tid = threadIdx.x;

  // ---- stage reflect-padded features for this row ----
  const float* fb = feat + (size_t)b * 64 * HW;
  for (int e = tid; e < 64 * 3 * 130; e += 256) {
    int ic = e / 390;
    int rem = e - ic * 390;
    int r = rem / 130;
    int wcol = rem - r * 130;                 // 0..129 -> w-1..w+128
    int hr = reflect_idx(h + r - 1, HH);
    int ws = reflect_idx(wcol - 1, WW);
    xs[((ic >> 1) * 3 + r) * 264 + wcol * 2 + (ic & 1)] = fb[ic * HW + hr * WW + ws];
  }
  __syncthreads();

  const int lane = tid & 31, wv = tid >> 5;
  const int p = lane & 15, hi = lane >> 4;
  const int w0 = wv * 16;

  const v2f* __restrict__ wf = (const v2f*)wp;   // packed A fragments (coalesced)
  const v2f* xs2 = (const v2f*)xs;

  v8f acc0 = {0.f, 0.f, 0.f, 0.f, 0.f, 0.f, 0.f, 0.f};  // channels  0..15
  v8f acc1 = acc0;                                       // channels 16..31 (18..31 zero wts)

  int kb = 0;
  for (int kpos = 0; kpos < 9; ++kpos) {
    const int dh = kpos / 3, dw = kpos - dh * 3;
    // B fragment: lane l, comp v -> X[k = 4*kb + 2*hi + v][pixel p]
    //   ic = 4*(kb&15) + 2*hi + v  (even base -> pair load), icp = 2*(kb&15)+hi
    const v2f* xrow = xs2 + (size_t)hi * 396 + dh * 132 + (w0 + p + dw);
#pragma unroll
    for (int j = 0; j < 16; ++j, ++kb) {
      v2f a0 = wf[kb * 64 + lane];
      v2f a1 = wf[kb * 64 + 32 + lane];
      v2f bx = xrow[(size_t)j * 792];          // (2j)*3*132 float2
      acc0 = __builtin_amdgcn_wmma_f32_16x16x4_f32(
          false, a0, false, bx, (short)0, acc0, false, false);
      acc1 = __builtin_amdgcn_wmma_f32_16x16x4_f32(
          false, a1, false, bx, (short)0, acc1, false, false);
    }
  }

  // ---- spill conv results: lane holds channels r+8*hi (tile0), 16+r+8*hi (tile1) of pixel p
  float* zrow = &zb[(wv * 16 + p) * 19];
#pragma unroll
  for (int r = 0; r < 8; ++r) zrow[r + 8 * hi] = acc0[r];
  if (hi == 0) {
    zrow[16] = acc1[0];
    zrow[17] = acc1[1];
  }
  __syncthreads();

  // ---- grouped softmax: lane (p, g=hi) normalizes conv channels {2k+g} over k=0..8
  const int g = hi;
  float z[9];
#pragma unroll
  for (int k = 0; k < 9; ++k) z[k] = zrow[2 * k + g] + bias[2 * k + g];
  float mx = z[0];
#pragma unroll
  for (int k = 1; k < 9; ++k) mx = fmaxf(mx, z[k]);
  float s = 0.0f;
#pragma unroll
  for (int k = 0; k < 9; ++k) {
    z[k] = expf(z[k] - mx);
    s += z[k];
  }
  const float inv = 1.0f / s;
  const int wpix = w0 + p;
#pragma unroll
  for (int k = 0; k < 9; ++k) {
    int ch = 2 * k + g;                        // ff channel (reference layout)
    int si = ch / 9, kk = ch - si * 9;         // consumer layout [s][k]
    ffw[(((size_t)b * 2 + si) * 9 + kk) * HW + h * WW + wpix] = z[k] * inv;
  }
}

// ---------------------------------------------------------------------------
// Kernel 2: out[b,c,h,w] = sum_k ffw[b][c&1][k][h][w] * inpt[b,c,reflect(h+dh-1),reflect(w+dw-1)]
// Pure memory-bound; coalesced along W; weight rows shared by 32 channels (L2).
// ---------------------------------------------------------------------------
__global__ __launch_bounds__(256) void combine(
    const float* __restrict__ inp, const float* __restrict__ ffw,
    float* __restrict__ out, int total) {
  int n = blockIdx.x * 256 + threadIdx.x;
  if (n >= total) return;
  int w = n & 127;
  int h = (n >> 7) & 127;
  int c = (n >> 14) & 63;
  int b = n >> 20;
  const float* xin = inp + (size_t)(b * 64 + c) * HW;
  const float* wg = ffw + (((size_t)b * 2 + (c & 1)) * 9) * HW + h * WW + w;
  float acc = 0.0f;
#pragma unroll
  for (int kpos = 0; kpos < 9; ++kpos) {
    int dh = kpos / 3, dw = kpos - dh * 3;
    int hr = reflect_idx(h + dh - 1, HH);
    int wr = reflect_idx(w + dw - 1, WW);
    acc = fmaf(wg[(size_t)kpos * HW], xin[hr * WW + wr], acc);
  }
  out[n] = acc;
}

// ---------------------------------------------------------------------------
extern "C" void kernel_launch(void* const* d_in, const int* in_sizes, int n_in,
                              void* d_out, int out_size, void* d_ws, size_t ws_size,
                              hipStream_t stream) {
  const float* features = (const float*)d_in[0];  // [8,64,128,128]
  const float* inpt     = (const float*)d_in[1];  // [8,64,128,128]
  const float* conv_w   = (const float*)d_in[2];  // [18,64,3,3]
  const float* conv_b   = (const float*)d_in[3];  // [18]
  float* out = (float*)d_out;

  // workspace: [0, 18432) packed A fragments; then ff weights [8][2][9][128][128]
  float* wp  = (float*)d_ws;
  float* ffw = wp + 18432;

  pack_w<<<(18432 + 255) / 256, 256, 0, stream>>>(conv_w, wp);
  conv_softmax<<<dim3(HH, 8), 256, 0, stream>>>(features, conv_b, wp, ffw);
  const int total = 8 * 64 * HW;
  combine<<<(total + 255) / 256, 256, 0, stream>>>(inpt, ffw, out, total);
}